// GatherRouter_4054449127995
// MI455X (gfx1250) — compile-verified
//
#include <hip/hip_runtime.h>
#include <stdint.h>

// ---------------------------------------------------------------------------
// GatherRouter (top-2 MoE combine): out[tag[p,t], :] += data[p,t, :]
// Tags per flow are permutations of [0,T) -> each output row gets exactly one
// row from flow0 and one from flow1. We invert the permutations, then do a
// conflict-free gather: out[t] = data[0, inv0[t]] + data[1, inv1[t]].
// Memory-bound: ~192 MiB total traffic -> ~8.6 us at 23.3 TB/s. WMMA is not
// justified (a selection-matrix GEMM would be ~275 GFLOP of fake work); the
// CDNA5-appropriate path is the async global->LDS DMA pipe.
// ---------------------------------------------------------------------------

// Kernel 1: invert the per-flow tag permutations.
// inv[p*T + tag] = t  (every slot written every call -> deterministic).
// PT == 2*T, so flow index is just a compare -- no integer division.
__global__ void gr_build_inverse(const long long* __restrict__ tags,
                                 int* __restrict__ inv, int PT, int T) {
    int i = blockIdx.x * blockDim.x + threadIdx.x;
    if (i < PT) {
        int tag = (int)tags[i];
        int p   = (i >= T) ? 1 : 0;     // P == 2 by construction
        int t   = i - (p ? T : 0);
        inv[p * T + tag] = t;
    }
}

// Kernel 2: gather the two source rows per output row via CDNA5 async DMA
// (global_load_async_to_lds_b128, ASYNCcnt-tracked), add, store. One block
// per output row; each lane moves 16 B per row per chunk. Each lane reads
// back only the LDS bytes it itself requested, so the per-wave
// s_wait_asynccnt is sufficient -- no workgroup barrier.
__global__ __launch_bounds__(512) void gr_gather_add(
        const float* __restrict__ data, const int* __restrict__ inv,
        float* __restrict__ out, int T, int D) {
    // 16 KiB: bytes [0 .. 8191] = flow0 chunk, [8192 .. 16383] = flow1 chunk
    __shared__ __align__(16) float lds[4096];

    const int t  = blockIdx.x;
    const int r0 = inv[t];          // row in flow0 carrying tag t
    const int r1 = inv[T + t];      // row in flow1 carrying tag t

    const float* row0 = data + (size_t)r0 * D;
    const float* row1 = data + ((size_t)T + (size_t)r1) * D;
    float*       orow = out  + (size_t)t  * D;

    const int      nvec = D >> 2;   // float4 elements per row
    const unsigned tid  = threadIdx.x;

    // Per-lane LDS byte offsets (low 32 bits of the generic address of a
    // __shared__ object are the LDS offset on gfx1250 flat addressing).
    const unsigned l0 = (unsigned)(uintptr_t)&lds[tid * 4u];
    const unsigned l1 = l0 + 8192u;
    const float4*  lp0 = (const float4*)&lds[tid * 4u];
    const float4*  lp1 = (const float4*)&lds[2048u + tid * 4u];

    for (int i = tid; i < nvec; i += blockDim.x) {
        uint64_t g0 = (uint64_t)(uintptr_t)(row0 + (size_t)i * 4);
        uint64_t g1 = (uint64_t)(uintptr_t)(row1 + (size_t)i * 4);

        // CDNA5 async DMA: 16 bytes per lane, global -> LDS.
        asm volatile("global_load_async_to_lds_b128 %0, %1, off"
                     :: "v"(l0), "v"(g0) : "memory");
        asm volatile("global_load_async_to_lds_b128 %0, %1, off"
                     :: "v"(l1), "v"(g1) : "memory");
        asm volatile("s_wait_asynccnt 0" ::: "memory");

        const float4 a = *lp0;
        const float4 b = *lp1;
        float4 c;
        c.x = a.x + b.x; c.y = a.y + b.y; c.z = a.z + b.z; c.w = a.w + b.w;
        ((float4*)orow)[i] = c;
    }
}

// ---------------------------------------------------------------------------
// Fallback (no scratch): two-pass scatter. Pass 0 stores flow0 rows, pass 1
// does a read-modify-write with flow1 rows. Conflict-free (permutations), but
// two dependent passes and ~320 MiB traffic, so only used if d_ws is tiny.
// ---------------------------------------------------------------------------
__global__ __launch_bounds__(512) void gr_scatter_row(
        const float* __restrict__ data, const long long* __restrict__ tags,
        float* __restrict__ out, int T, int D, int flow) {
    const int t   = blockIdx.x;
    const int row = (int)tags[(size_t)flow * T + t];
    const float4* src = (const float4*)(data + ((size_t)flow * T + t) * (size_t)D);
    float4*       dst = (float4*)(out + (size_t)row * D);
    const int nvec = D >> 2;
    for (int i = threadIdx.x; i < nvec; i += blockDim.x) {
        float4 v = src[i];
        if (flow != 0) {
            float4 o = dst[i];
            v.x += o.x; v.y += o.y; v.z += o.z; v.w += o.w;
        }
        dst[i] = v;
    }
}

extern "C" void kernel_launch(void* const* d_in, const int* in_sizes, int n_in,
                              void* d_out, int out_size, void* d_ws, size_t ws_size,
                              hipStream_t stream) {
    const float*     data = (const float*)d_in[0];
    const long long* tags = (const long long*)d_in[1];
    float*           out  = (float*)d_out;

    const int PT = in_sizes[1];      // P*T (P == 2 by construction)
    const int T  = PT / 2;
    const int D  = in_sizes[0] / PT;

    if (ws_size >= (size_t)PT * sizeof(int)) {
        int* inv = (int*)d_ws;
        gr_build_inverse<<<(PT + 255) / 256, 256, 0, stream>>>(tags, inv, PT, T);
        gr_gather_add<<<T, 512, 0, stream>>>(data, inv, out, T, D);
    } else {
        gr_scatter_row<<<T, 512, 0, stream>>>(data, tags, out, T, D, 0);
        gr_scatter_row<<<T, 512, 0, stream>>>(data, tags, out, T, D, 1);
    }
}